// ComplexMixture_54838142435828
// MI455X (gfx1250) — compile-verified
//
#include <hip/hip_runtime.h>
#include <stdint.h>

// CDNA5 / gfx1250 — wave32, WMMA F32 16x16x4 + Tensor Data Mover staging.
typedef __attribute__((ext_vector_type(2))) float v2f;
typedef __attribute__((ext_vector_type(8))) float v8f;
typedef __attribute__((ext_vector_type(4))) unsigned int v4u;
typedef __attribute__((ext_vector_type(8))) int v8i;
typedef __attribute__((ext_vector_type(4))) int v4i;

#define B_    64
#define T_    256
#define D_    512
#define KTOT  512          // stacked K = 2*T
#define BK    32           // k-slab staged in LDS
#define BM    128          // block tile M (d index)
#define BN    64           // block tile N (e index)
#define LDSWA 132          // A/S tile LDS row stride: 128 + 4 (TDM pad @128dw)
#define LDSWY 68           // Y   tile LDS row stride:  64 + 4 (TDM pad @64dw)
#define NSLAB (KTOT / BK)  // 16

// Issue one TDM 2-D tile DMA: 32 rows x tile_w floats, row stride 512 floats,
// LDS padding: +4 DWORDs after every (2^(pad_code+1)) DWORDs (set = tile_w).
__device__ __forceinline__ void tdm_tile_load(const float* gsrc, void* ldst,
                                              unsigned tile_w, unsigned pad_code) {
    const uint64_t ga   = (uint64_t)(uintptr_t)gsrc;
    const uint32_t loff = (uint32_t)(uintptr_t)ldst;   // low 32 bits = LDS byte offset

    v4u g0;
    g0.x = 1u;                                              // count=1 (valid user D#)
    g0.y = loff;                                            // lds_addr
    g0.z = (uint32_t)ga;                                    // global_addr[31:0]
    g0.w = ((uint32_t)(ga >> 32) & 0x01FFFFFFu) | (2u << 30); // global_addr[56:32] | type=2

    v8i g1;
    g1[0] = (int)((2u << 16)          // data_size = 4 bytes
                | (1u << 20)          // pad_enable
                | (pad_code << 22)    // pad_interval: 2^(code+1) DWORDs
                | (3u << 25));        // pad_amount:   4 DWORDs
    g1[1] = (int)(512u << 16);        // tensor_dim0 = 512 (row length, elements)
    g1[2] = (int)(256u << 16);        // tensor_dim1 = 256 (rows)
    g1[3] = (int)(tile_w << 16);      // tile_dim0
    g1[4] = 32;                       // tile_dim1 = 32
    g1[5] = 512;                      // tensor_dim0_stride = 512 elements
    g1[6] = 0;
    g1[7] = 0;

    v4i z4 = {0, 0, 0, 0};                    // 2-D tensor: groups 2/3 unused
    v8i z8 = {0, 0, 0, 0, 0, 0, 0, 0};        // 6-arg form: extra group unused
    __builtin_amdgcn_tensor_load_to_lds(g0, g1, z4, z4, z8, 0);
}

__global__ __launch_bounds__(256)
void complex_mixture_wmma_tdm(const float* __restrict__ real,
                              const float* __restrict__ imag,
                              const float* __restrict__ weight,
                              float* __restrict__ out)
{
    __shared__ float sA[2][BK][LDSWA];   // A(out_r) source: X[k, m]      (unscaled)
    __shared__ float sS[2][BK][LDSWA];   // A(out_i) source: X[k^256, m]  (unscaled)
    __shared__ float sY[2][BK][LDSWY];   // B operand: X[k, n]
    __shared__ float sWs[KTOT];          // signed weights: (kg < T ? +w[t] : -w[t])

    const int b  = blockIdx.z;
    const int m0 = blockIdx.y * BM;
    const int n0 = blockIdx.x * BN;

    const int tid  = threadIdx.x;
    const int lane = tid & 31;        // wave32
    const int wave = tid >> 5;        // 8 waves
    const int l15  = lane & 15;
    const int kh   = lane >> 4;       // K-half select per WMMA fragment layout

    const int wr = wave & 3;          // wave M strip: rows wr*32 .. wr*32+31
    const int wc = wave >> 2;         // wave N strip: cols wc*32 .. wc*32+31

    const float* base_r = real + (size_t)b * T_ * D_;
    const float* base_i = imag + (size_t)b * T_ * D_;
    const float* wb     = weight + (size_t)b * T_;

    // Stage signed weight table (256 threads cover 512 entries).
    {
        const float w = wb[tid];
        sWs[tid]      =  w;
        sWs[T_ + tid] = -w;
    }

    // Prime the pipeline: slab 0 -> buffer 0 (wave 0 drives the TDM).
    if (wave == 0) {
        tdm_tile_load(base_r + m0, &sA[0][0][0], BM, 6);   // slab0 low half: A from real
        tdm_tile_load(base_i + m0, &sS[0][0][0], BM, 6);   // swapped from imag
        tdm_tile_load(base_r + n0, &sY[0][0][0], BN, 5);   // B from real
    }

    v8f acc_r00 = {}; v8f acc_r01 = {}; v8f acc_r10 = {}; v8f acc_r11 = {};
    v8f acc_i00 = {}; v8f acc_i01 = {}; v8f acc_i10 = {}; v8f acc_i11 = {};

    const int am0 = wr * 32 + l15;    // A fragment rows (local m), two subtiles
    const int am1 = am0 + 16;
    const int bn0 = wc * 32 + l15;    // B fragment cols (local n), two subtiles
    const int bn1 = bn0 + 16;

    for (int s = 0; s < NSLAB; ++s) {
        const int cur = s & 1;

        if (wave == 0) {
            if (s + 1 < NSLAB) {
                const int   ks2  = (s + 1) * BK;
                const int   t2   = ks2 & (T_ - 1);
                const bool  low2 = (ks2 < T_);
                const float* p0  = (low2 ? base_r : base_i) + (size_t)t2 * D_;
                const float* p1  = (low2 ? base_i : base_r) + (size_t)t2 * D_;
                const int nb = 1 - cur;
                tdm_tile_load(p0 + m0, &sA[nb][0][0], BM, 6);
                tdm_tile_load(p1 + m0, &sS[nb][0][0], BM, 6);
                tdm_tile_load(p0 + n0, &sY[nb][0][0], BN, 5);
                __builtin_amdgcn_s_wait_tensorcnt(3);   // slab s's 3 DMAs done (in-order)
            } else {
                __builtin_amdgcn_s_wait_tensorcnt(0);   // drain tail
            }
        }
        __syncthreads();

        const float (*tA)[LDSWA] = sA[cur];
        const float (*tS)[LDSWA] = sS[cur];
        const float (*tY)[LDSWY] = sY[cur];
        const int kw = s * BK;        // stacked-k base for the weight table

#pragma unroll
        for (int k4 = 0; k4 < BK; k4 += 4) {
            // A 16x4: lanes 0-15 -> K={k4,k4+1}, lanes 16-31 -> K={k4+2,k4+3}
            const int kr = k4 + kh * 2;

            const float w0 = sWs[kw + kr];       // signed weight for rows kr, kr+1
            const float w1 = sWs[kw + kr + 1];
            const float a0 = __builtin_fabsf(w0);
            const float a1 = __builtin_fabsf(w1);

            v2f aw0; aw0.x = tA[kr][am0] * a0;  aw0.y = tA[kr + 1][am0] * a1;
            v2f aw1; aw1.x = tA[kr][am1] * a0;  aw1.y = tA[kr + 1][am1] * a1;
            v2f as0; as0.x = tS[kr][am0] * w0;  as0.y = tS[kr + 1][am0] * w1;
            v2f as1; as1.x = tS[kr][am1] * w0;  as1.y = tS[kr + 1][am1] * w1;
            v2f b0;  b0.x  = tY[kr][bn0];       b0.y  = tY[kr + 1][bn0];
            v2f b1;  b1.x  = tY[kr][bn1];       b1.y  = tY[kr + 1][bn1];

            acc_r00 = __builtin_amdgcn_wmma_f32_16x16x4_f32(false, aw0, false, b0, (short)0, acc_r00, false, false);
            acc_r01 = __builtin_amdgcn_wmma_f32_16x16x4_f32(false, aw0, false, b1, (short)0, acc_r01, false, false);
            acc_r10 = __builtin_amdgcn_wmma_f32_16x16x4_f32(false, aw1, false, b0, (short)0, acc_r10, false, false);
            acc_r11 = __builtin_amdgcn_wmma_f32_16x16x4_f32(false, aw1, false, b1, (short)0, acc_r11, false, false);
            acc_i00 = __builtin_amdgcn_wmma_f32_16x16x4_f32(false, as0, false, b0, (short)0, acc_i00, false, false);
            acc_i01 = __builtin_amdgcn_wmma_f32_16x16x4_f32(false, as0, false, b1, (short)0, acc_i01, false, false);
            acc_i10 = __builtin_amdgcn_wmma_f32_16x16x4_f32(false, as1, false, b0, (short)0, acc_i10, false, false);
            acc_i11 = __builtin_amdgcn_wmma_f32_16x16x4_f32(false, as1, false, b1, (short)0, acc_i11, false, false);
        }

        __syncthreads();   // everyone done reading buffer `cur` before it is refilled
    }

    // Epilogue: C/D layout -> lanes 0-15: M=v, lanes 16-31: M=8+v; N=l15.
    const size_t outr_base = (size_t)b * D_ * D_;
    const size_t outi_base = (size_t)B_ * D_ * D_ + outr_base;
    const int gm  = m0 + wr * 32 + kh * 8;    // M-subtile 0; subtile 1 at +16
    const int gn0 = n0 + wc * 32 + l15;
    const int gn1 = gn0 + 16;

#pragma unroll
    for (int v = 0; v < 8; ++v) {
        const size_t ro0 = (size_t)(gm + v) * D_;
        const size_t ro1 = (size_t)(gm + 16 + v) * D_;
        out[outr_base + ro0 + gn0] = acc_r00[v];
        out[outr_base + ro0 + gn1] = acc_r01[v];
        out[outr_base + ro1 + gn0] = acc_r10[v];
        out[outr_base + ro1 + gn1] = acc_r11[v];
        out[outi_base + ro0 + gn0] = acc_i00[v];
        out[outi_base + ro0 + gn1] = acc_i01[v];
        out[outi_base + ro1 + gn0] = acc_i10[v];
        out[outi_base + ro1 + gn1] = acc_i11[v];
    }
}

extern "C" void kernel_launch(void* const* d_in, const int* in_sizes, int n_in,
                              void* d_out, int out_size, void* d_ws, size_t ws_size,
                              hipStream_t stream) {
    (void)in_sizes; (void)n_in; (void)out_size; (void)d_ws; (void)ws_size;
    const float* real   = (const float*)d_in[0];
    const float* imag   = (const float*)d_in[1];
    const float* weight = (const float*)d_in[2];
    float*       out    = (float*)d_out;

    dim3 grid(D_ / BN, D_ / BM, B_);   // 8 x 4 x 64
    dim3 block(256);                   // 8 wave32s
    complex_mixture_wmma_tdm<<<grid, block, 0, stream>>>(real, imag, weight, out);
}